// RNNAttentionDecoder_54391465837288
// MI455X (gfx1250) — compile-verified
//
#include <hip/hip_runtime.h>
#include <hip/hip_bf16.h>
#include <math.h>

// Problem dims (from reference)
#define I_SZ 1024
#define H_SZ 1024
#define V_SZ 50257
#define S_SZ 4096
#define G_SZ (4 * H_SZ)   // 4096 gate rows
#define H2_SZ (2 * H_SZ)  // 2048 concat length

#define ATT_SBLK 64       // s-chunks for attention partial reduction

typedef float v2f __attribute__((ext_vector_type(2)));
typedef float v8f __attribute__((ext_vector_type(8)));

#if defined(__has_builtin)
#if __has_builtin(__builtin_amdgcn_wmma_f32_16x16x4_f32)
#define HAVE_WMMA_F32_16X16X4 1
#endif
#endif
#ifndef HAVE_WMMA_F32_16X16X4
#define HAVE_WMMA_F32_16X16X4 0
#endif

// ---------------------------------------------------------------------------
// WMMA-based GEMV:  out[r] = dot(M[r,0:K], vec) (+ dot(M2[r],vec2))
//                            (+ bias[r]) (+ bias2[r])
// One wave computes 16 rows with V_WMMA_F32_16X16X4_F32; block = 8 waves =
// 128 rows. The vector(s) live in LDS.
//
// K-slot remap trick: inside one WMMA the mapping of K-slots to memory
// columns is arbitrary as long as A and B agree. We let each lane load a
// contiguous float4 of its row:
//   lane (m, kh=0): columns kk+0..kk+3     lane (m, kh=1): columns kk+4..kk+7
// and feed TWO WMMAs from that single b128 load:
//   WMMA#1: A slots (0,1,2,3) = cols (kk+0, kk+1, kk+4, kk+5)
//   WMMA#2: A slots (0,1,2,3) = cols (kk+2, kk+3, kk+6, kk+7)
// with B (broadcast across all 16 N columns) permuted identically from a
// single ds b128 read:  bv = sv[kk+4*kh .. kk+4*kh+3].
// Lanes (m, m+16) jointly read 32 contiguous bytes of row m -> each
// global_load_b128 moves 512 coalesced bytes per wave.
// Every C column equals the dot product, so extraction only relies on the
// documented C layout (lane 0: M=0..7 in c[0..7]; lane 16: M=8..15).
// ---------------------------------------------------------------------------
__global__ __launch_bounds__(256) void gemv16_wmma(
    const float* __restrict__ M,    const float* __restrict__ vec,
    const float* __restrict__ M2,   const float* __restrict__ vec2,
    const float* __restrict__ bias, const float* __restrict__ bias2,
    float* __restrict__ out, int rows, int K)
{
    __shared__ __align__(16) float sv[H2_SZ];   // up to 2048 floats (8 KB)

    const int tid = threadIdx.x;
    for (int i = tid; i < K; i += 256) sv[i] = vec[i];
    if (M2 != nullptr) {
        // second vector stored behind the first (only used when K == 1024)
        for (int i = tid; i < K; i += 256) sv[K + i] = vec2[i];
    }
    __syncthreads();

    const int lane = tid & 31;       // wave32 lane id
    const int wave = tid >> 5;
    const int m    = lane & 15;      // row within the 16-row tile
    const int kh   = lane >> 4;      // lane half: selects column group
    const int row0 = blockIdx.x * 128 + wave * 16;
    if (row0 >= rows) return;        // wave-uniform exit (EXEC stays full)

    const int r  = row0 + m;
    const int lr = (r < rows) ? r : (rows - 1);   // clamp loads for tail rows
    const float* __restrict__ Mr = M + (size_t)lr * (size_t)K;

#if HAVE_WMMA_F32_16X16X4
    v8f c = {0.f, 0.f, 0.f, 0.f, 0.f, 0.f, 0.f, 0.f};
#else
    float acc = 0.f;
#endif

#pragma unroll 4
    for (int kk = 0; kk < K; kk += 8) {
        const int ko = kk + 4 * kh;
        const float4 av = *reinterpret_cast<const float4*>(Mr + ko);
        const float4 bv = *reinterpret_cast<const float4*>(sv + ko);
#if HAVE_WMMA_F32_16X16X4
        v2f a1; a1.x = av.x; a1.y = av.y;
        v2f b1; b1.x = bv.x; b1.y = bv.y;
        v2f a2; a2.x = av.z; a2.y = av.w;
        v2f b2; b2.x = bv.z; b2.y = bv.w;
        c = __builtin_amdgcn_wmma_f32_16x16x4_f32(
                false, a1, false, b1, (short)0, c, false, false);
        c = __builtin_amdgcn_wmma_f32_16x16x4_f32(
                false, a2, false, b2, (short)0, c, false, false);
#else
        acc = fmaf(av.x, bv.x, fmaf(av.y, bv.y,
              fmaf(av.z, bv.z, fmaf(av.w, bv.w, acc))));
#endif
    }

    if (M2 != nullptr) {
        const float* __restrict__ Mr2 = M2 + (size_t)lr * (size_t)K;
#pragma unroll 4
        for (int kk = 0; kk < K; kk += 8) {
            const int ko = kk + 4 * kh;
            const float4 av = *reinterpret_cast<const float4*>(Mr2 + ko);
            const float4 bv = *reinterpret_cast<const float4*>(sv + K + ko);
#if HAVE_WMMA_F32_16X16X4
            v2f a1; a1.x = av.x; a1.y = av.y;
            v2f b1; b1.x = bv.x; b1.y = bv.y;
            v2f a2; a2.x = av.z; a2.y = av.w;
            v2f b2; b2.x = bv.z; b2.y = bv.w;
            c = __builtin_amdgcn_wmma_f32_16x16x4_f32(
                    false, a1, false, b1, (short)0, c, false, false);
            c = __builtin_amdgcn_wmma_f32_16x16x4_f32(
                    false, a2, false, b2, (short)0, c, false, false);
#else
            acc = fmaf(av.x, bv.x, fmaf(av.y, bv.y,
                  fmaf(av.z, bv.z, fmaf(av.w, bv.w, acc))));
#endif
        }
    }

#if HAVE_WMMA_F32_16X16X4
    // C layout: lane 0 holds rows M=0..7 in c[0..7]; lane 16 holds M=8..15.
    if (lane == 0 || lane == 16) {
        const int base = row0 + ((lane == 16) ? 8 : 0);
#pragma unroll
        for (int j = 0; j < 8; ++j) {
            const int rr = base + j;
            if (rr < rows) {
                float v = c[j];
                if (bias)  v += bias[rr];
                if (bias2) v += bias2[rr];
                out[rr] = v;
            }
        }
    }
#else
    // Fallback: combine the two column-group partials of row m across halves.
    const float other = __shfl_xor(acc, 16, 32);
    const float total = acc + other;
    if (kh == 0 && r < rows) {
        float v = total;
        if (bias)  v += bias[r];
        if (bias2) v += bias2[r];
        out[r] = v;
    }
#endif
}

// ---------------------------------------------------------------------------
// LSTM pointwise: gates (torch order i,f,g,o) -> c_new, h_new
// ---------------------------------------------------------------------------
__device__ __forceinline__ float sigmoidf(float x) {
    return 1.0f / (1.0f + expf(-x));
}

__global__ __launch_bounds__(256) void lstm_pointwise(
    const float* __restrict__ gates, const float* __restrict__ c_prev,
    float* __restrict__ h_out, float* __restrict__ c_out,
    float* __restrict__ h_copy_ws)
{
    const int j = blockIdx.x * 256 + threadIdx.x;
    if (j >= H_SZ) return;
    const float ig = sigmoidf(gates[j]);
    const float fg = sigmoidf(gates[j + H_SZ]);
    const float gg = tanhf(gates[j + 2 * H_SZ]);
    const float og = sigmoidf(gates[j + 3 * H_SZ]);
    const float cn = fg * c_prev[j] + ig * gg;
    const float hn = og * tanhf(cn);
    c_out[j] = cn;
    h_out[j] = hn;
    h_copy_ws[j] = hn;   // front half of concat buffer
}

// ---------------------------------------------------------------------------
// Softmax over 4096 scores (single workgroup of 1024 threads, 4 elems each)
// ---------------------------------------------------------------------------
__global__ __launch_bounds__(1024) void softmax4096(
    const float* __restrict__ ratios, float* __restrict__ prob)
{
    __shared__ float red[1024];
    const int tid = threadIdx.x;
    const float4 r = reinterpret_cast<const float4*>(ratios)[tid];

    float lm = fmaxf(fmaxf(r.x, r.y), fmaxf(r.z, r.w));
    red[tid] = lm;
    __syncthreads();
    for (int s = 512; s > 0; s >>= 1) {
        if (tid < s) red[tid] = fmaxf(red[tid], red[tid + s]);
        __syncthreads();
    }
    const float gmax = red[0];
    __syncthreads();

    float4 e;
    e.x = expf(r.x - gmax); e.y = expf(r.y - gmax);
    e.z = expf(r.z - gmax); e.w = expf(r.w - gmax);
    red[tid] = e.x + e.y + e.z + e.w;
    __syncthreads();
    for (int s = 512; s > 0; s >>= 1) {
        if (tid < s) red[tid] += red[tid + s];
        __syncthreads();
    }
    const float inv = 1.0f / red[0];

    float4 p; p.x = e.x * inv; p.y = e.y * inv; p.z = e.z * inv; p.w = e.w * inv;
    reinterpret_cast<float4*>(prob)[tid] = p;
}

// ---------------------------------------------------------------------------
// attention = enc^T @ prob, split over ATT_SBLK s-chunks for parallelism.
// Block b handles s in [b*64, b*64+64); thread t owns columns 4t..4t+3 with
// float4 (b128) loads; prob[s] is wave-uniform (scalar broadcast).
// ---------------------------------------------------------------------------
__global__ __launch_bounds__(256) void attn_partial(
    const float* __restrict__ enc, const float* __restrict__ prob,
    float* __restrict__ part)   // [ATT_SBLK][H_SZ]
{
    const int t  = threadIdx.x;                 // column group: 4t..4t+3
    const int s0 = blockIdx.x * (S_SZ / ATT_SBLK);
    const float4* __restrict__ enc4 = reinterpret_cast<const float4*>(enc);

    float4 acc; acc.x = 0.f; acc.y = 0.f; acc.z = 0.f; acc.w = 0.f;
#pragma unroll 4
    for (int i = 0; i < S_SZ / ATT_SBLK; ++i) {
        const int s   = s0 + i;
        const float p = prob[s];                // uniform -> s_load broadcast
        const float4 e = enc4[(size_t)s * (H_SZ / 4) + t];
        acc.x = fmaf(e.x, p, acc.x);
        acc.y = fmaf(e.y, p, acc.y);
        acc.z = fmaf(e.z, p, acc.z);
        acc.w = fmaf(e.w, p, acc.w);
    }
    reinterpret_cast<float4*>(part)[(size_t)blockIdx.x * (H_SZ / 4) + t] = acc;
}

__global__ __launch_bounds__(256) void attn_final(
    const float* __restrict__ part, float* __restrict__ attn_out)
{
    const int j = blockIdx.x * 256 + threadIdx.x;   // 4 blocks cover 1024 cols
    float acc = 0.f;
#pragma unroll 8
    for (int b = 0; b < ATT_SBLK; ++b) acc += part[(size_t)b * H_SZ + j];
    attn_out[j] = acc;
}

// ---------------------------------------------------------------------------
// Launch: gates GEMV -> LSTM pointwise -> scores GEMV -> softmax ->
//         attention partial+final -> output-projection GEMV (411 MB, dominant)
// ---------------------------------------------------------------------------
extern "C" void kernel_launch(void* const* d_in, const int* in_sizes, int n_in,
                              void* d_out, int out_size, void* d_ws, size_t ws_size,
                              hipStream_t stream)
{
    (void)in_sizes; (void)n_in; (void)out_size; (void)ws_size;

    const float* x     = (const float*)d_in[0];   // [I]
    const float* h     = (const float*)d_in[1];   // [H]
    const float* c     = (const float*)d_in[2];   // [H]
    // d_in[3] = dummy (int)
    const float* enc   = (const float*)d_in[4];   // [S,H]
    const float* W_ih  = (const float*)d_in[5];   // [4H,I]
    const float* W_hh  = (const float*)d_in[6];   // [4H,H]
    const float* b_ih  = (const float*)d_in[7];   // [4H]
    const float* b_hh  = (const float*)d_in[8];   // [4H]
    const float* W_out = (const float*)d_in[9];   // [V,2H]
    const float* b_out = (const float*)d_in[10];  // [V]

    float* out = (float*)d_out;               // [logits V | h_new H | c_new H]
    float* ws  = (float*)d_ws;
    float* ws_gates  = ws;                    // 4096
    float* ws_ratios = ws + G_SZ;             // 4096
    float* ws_prob   = ws + 2 * G_SZ;         // 4096
    float* ws_concat = ws + 3 * G_SZ;         // 2048 = [h_new | attention]
    float* ws_part   = ws + 3 * G_SZ + H2_SZ; // ATT_SBLK*1024 floats (256 KB)

    // 1) gates = W_ih@x + W_hh@h + b_ih + b_hh   (4096 rows, K=1024 each)
    gemv16_wmma<<<G_SZ / 128, 256, 0, stream>>>(
        W_ih, x, W_hh, h, b_ih, b_hh, ws_gates, G_SZ, I_SZ);

    // 2) LSTM cell pointwise -> h_new (out+V, ws_concat[0:H]), c_new (out+V+H)
    lstm_pointwise<<<H_SZ / 256, 256, 0, stream>>>(
        ws_gates, c, out + V_SZ, out + V_SZ + H_SZ, ws_concat);

    // 3) ratios[s] = enc[s,:] . h_new   (4096 rows, K=1024)
    gemv16_wmma<<<S_SZ / 128, 256, 0, stream>>>(
        enc, ws_concat, nullptr, nullptr, nullptr, nullptr,
        ws_ratios, S_SZ, H_SZ);

    // 4) prob = softmax(ratios)
    softmax4096<<<1, 1024, 0, stream>>>(ws_ratios, ws_prob);

    // 5) attention = enc^T @ prob  -> ws_concat[H:2H]
    attn_partial<<<ATT_SBLK, 256, 0, stream>>>(enc, ws_prob, ws_part);
    attn_final<<<H_SZ / 256, 256, 0, stream>>>(ws_part, ws_concat + H_SZ);

    // 6) logits = W_out @ concat + b_out   (50257 rows, K=2048) — dominant
    const int blocksF = (V_SZ + 127) / 128;   // 393
    gemv16_wmma<<<blocksF, 256, 0, stream>>>(
        W_out, ws_concat, nullptr, nullptr, b_out, nullptr,
        out, V_SZ, H2_SZ);
}